// VAErModel_77464030151294
// MI455X (gfx1250) — compile-verified
//
#include <hip/hip_runtime.h>
#include <math.h>

// ---------------------------------------------------------------------------
// Problem constants (match the JAX reference)
// ---------------------------------------------------------------------------
#define B_DIM 64
#define S_DIM 512
#define V_DIM 2000
#define NVEC4 (V_DIM / 4)      // 500 float4 per row (8000 B rows -> 16B aligned)

typedef float v2f __attribute__((ext_vector_type(2)));
typedef float v4f __attribute__((ext_vector_type(4)));
typedef float v8f __attribute__((ext_vector_type(8)));

// ---------------------------------------------------------------------------
// ws[0] = cro_loss accumulator, ws[1] = sum_b (sum_s pred)/L accumulator
// ---------------------------------------------------------------------------
__global__ void VAEr_zero_ws(float* __restrict__ ws) {
    ws[0] = 0.0f;
    ws[1] = 0.0f;
}

// ---------------------------------------------------------------------------
// One 8-wave workgroup per (b, s) row. Dead rows (s >= L) exit before any
// logit traffic. Live rows stream 2000 f32 once with NT-hinted b128 loads,
// do an online (max, sum-exp) + label-gather reduction, then two f32 global
// atomics. Memory-bound by design: ~134 MB effective / 23.3 TB/s.
// ---------------------------------------------------------------------------
__global__ __launch_bounds__(256) void VAEr_row_ce_kernel(
    const float* __restrict__ outputs,   // [B,S,V]
    const int*   __restrict__ label,     // [B,S]
    const int*   __restrict__ seqlen,    // [B]
    float*       __restrict__ ws)
{
    const int row = blockIdx.x;              // 0 .. B*S-1
    const int b   = row / S_DIM;
    const int s   = row - b * S_DIM;
    const int L   = seqlen[b] - 2;           // effective length, >= 14
    if (s >= L) return;                      // masked row: skip the 8KB read

    const int t = threadIdx.x;
    const v4f* rowv = (const v4f*)(outputs + (size_t)row * V_DIM);

    // --- load up to 2 float4 per thread (500 vectors / 256 threads) ---------
    v4f a0 = __builtin_nontemporal_load(rowv + t);
    const bool has2 = (t + 256) < NVEC4;     // threads 0..243
    v4f a1;
    if (has2) a1 = __builtin_nontemporal_load(rowv + t + 256);
    else      a1 = (v4f){-INFINITY, -INFINITY, -INFINITY, -INFINITY};

    // --- label logit gather (exactly one thread in the block hits) ---------
    const int lab = label[row];
    const int lv = lab >> 2, lc = lab & 3;
    float p = 0.0f;
    if (t == lv)                p = a0[lc];
    else if (has2 && (t + 256) == lv) p = a1[lc];

    // --- thread-local max + sum-exp -----------------------------------------
    float m = fmaxf(fmaxf(a0.x, a0.y), fmaxf(a0.z, a0.w));
    m = fmaxf(m, fmaxf(fmaxf(a1.x, a1.y), fmaxf(a1.z, a1.w)));
    float sum = __expf(a0.x - m) + __expf(a0.y - m) +
                __expf(a0.z - m) + __expf(a0.w - m);
    // exp(-inf - m) == 0 for the tail lanes, so no masking needed:
    sum += __expf(a1.x - m) + __expf(a1.y - m) +
           __expf(a1.z - m) + __expf(a1.w - m);

    // --- wave32 reduction: merge (m, sum), add p ----------------------------
    #pragma unroll
    for (int off = 16; off > 0; off >>= 1) {
        const float m2 = __shfl_xor(m,   off, 32);
        const float s2 = __shfl_xor(sum, off, 32);
        const float p2 = __shfl_xor(p,   off, 32);
        const float mn = fmaxf(m, m2);
        sum = sum * __expf(m - mn) + s2 * __expf(m2 - mn);
        m = mn;
        p += p2;
    }

    // --- cross-wave merge through LDS (8 waves) -----------------------------
    __shared__ float lm[8], ls[8], lp[8];
    const int wave = t >> 5;
    if ((t & 31) == 0) { lm[wave] = m; ls[wave] = sum; lp[wave] = p; }
    __syncthreads();

    if (t == 0) {
        float M = lm[0], S = ls[0], P = lp[0];
        #pragma unroll
        for (int w = 1; w < 8; ++w) {
            const float mn = fmaxf(M, lm[w]);
            S = S * __expf(M - mn) + ls[w] * __expf(lm[w] - mn);
            M = mn;
            P += lp[w];
        }
        const float nll = (M + __logf(S)) - P;     // -log_softmax[label]
        const float Lf  = (float)L;
        atomicAdd(&ws[0], nll / (Lf * Lf));        // -> cro_loss
        atomicAdd(&ws[1], P / Lf);                 // -> sum_b (sum pred)/L
    }
}

// ---------------------------------------------------------------------------
// Final scalar kernel, one full wave32 (EXEC all-ones as WMMA requires).
// The 64-element gaussian-loss mean is reduced through V_WMMA_F32_16X16X4_F32
// with an all-ones B: every column of D equals A's row sums, so
// sum(lane's 8 D regs) + shfl_xor(.,16) == total sum of A's 64 elements,
// independent of the A VGPR layout (sum is permutation-invariant).
// ---------------------------------------------------------------------------
__global__ __launch_bounds__(32) void VAEr_final_kernel(
    const float* __restrict__ gauss,     // [64]
    const float* __restrict__ uni,       // [1]
    const float* __restrict__ ws,        // [2]
    float*       __restrict__ out)       // [3] = (loss, rec_loss, avg_pre_pro)
{
    const int lane = threadIdx.x;

    v2f A;  A.x = gauss[lane];  A.y = gauss[lane + 32];   // 64 values into 16x4 A
    v2f Bm; Bm.x = 1.0f;        Bm.y = 1.0f;              // all-ones 4x16 B
    v8f C = {};
    // 8 args: (neg_a, A, neg_b, B, c_mod, C, reuse_a, reuse_b)
    v8f D = __builtin_amdgcn_wmma_f32_16x16x4_f32(
        /*neg_a=*/false, A, /*neg_b=*/false, Bm,
        /*c_mod=*/(short)0, C, /*reuse_a=*/false, /*reuse_b=*/false);

    float part = D[0] + D[1] + D[2] + D[3] + D[4] + D[5] + D[6] + D[7];
    part += __shfl_xor(part, 16, 32);            // lo-rows + hi-rows = full sum
    const float gmean = part * (1.0f / 64.0f);

    const float rec = ws[0] * (1.0f / (float)B_DIM);   // cro_loss / B
    const float avg = ws[1] * (1.0f / (float)B_DIM);   // mean over b
    const float u   = uni[0];

    if (lane == 0) {
        out[0] = 1e-05f * rec + (100.0f / 256.0f) * gmean + 10.0f * u;
        out[1] = rec;
        out[2] = avg;
    }
}

// ---------------------------------------------------------------------------
extern "C" void kernel_launch(void* const* d_in, const int* in_sizes, int n_in,
                              void* d_out, int out_size, void* d_ws, size_t ws_size,
                              hipStream_t stream) {
    (void)in_sizes; (void)n_in; (void)out_size; (void)ws_size;
    const float* outputs = (const float*)d_in[0];   // [B,S,V] f32
    const int*   label   = (const int*)  d_in[1];   // [B,S]   i32
    const int*   seqlen  = (const int*)  d_in[2];   // [B]     i32
    const float* gauss   = (const float*)d_in[3];   // [B]     f32
    const float* uni     = (const float*)d_in[4];   // [1]     f32
    float* ws  = (float*)d_ws;
    float* out = (float*)d_out;

    VAEr_zero_ws<<<1, 1, 0, stream>>>(ws);
    VAEr_row_ce_kernel<<<B_DIM * S_DIM, 256, 0, stream>>>(outputs, label, seqlen, ws);
    VAEr_final_kernel<<<1, 32, 0, stream>>>(gauss, uni, ws, out);
}